// KRR_1632087572587
// MI455X (gfx1250) — compile-verified
//
#include <hip/hip_runtime.h>

typedef __attribute__((ext_vector_type(2))) float v2f;
typedef __attribute__((ext_vector_type(8))) float v8f;

#define NTEST   16384
#define NTRAIN  8192
#define DIM     64
#define NSPLIT  4
#define WPB     8                       // waves per block; each wave owns one 16-row test tile
#define NB      8                       // 16-row train tiles staged per round (= 128 rows/round)
#define ROWS_PER_ROUND (NB * 16)
#define TILE_FLOATS (16 * 64)           // one staged tile: 16 kk-groups * 32 lanes * 2 floats
#define TILE_BYTES  (TILE_FLOATS * 4)   // 4 KB
#define BUF_FLOATS  (NB * TILE_FLOATS)  // 8192 floats = 32 KB
#define BUF_BYTES   (BUF_FLOATS * 4)

// ---------------------------------------------------------------- row norms
__global__ __launch_bounds__(256) void krr_norms_kernel(
    const float* __restrict__ X, float* __restrict__ out, int nrows) {
  int r = blockIdx.x * blockDim.x + threadIdx.x;
  if (r >= nrows) return;
  const float4* p = (const float4*)(X + (size_t)r * DIM);
  float s = 0.f;
#pragma unroll
  for (int i = 0; i < DIM / 4; ++i) {
    float4 v = p[i];
    s = fmaf(v.x, v.x, s); s = fmaf(v.y, v.y, s);
    s = fmaf(v.z, v.z, s); s = fmaf(v.w, v.w, s);
  }
  out[r] = s;
}

// Issue 16 async global->LDS b64 copies that land one 16x64 train tile in LDS
// pre-swizzled into the WMMA B lane order: element (kk, lane) at (kk*32+lane)*8.
// Tracked by ASYNCcnt; caller must s_wait_asynccnt + barrier before consuming.
__device__ __forceinline__ void stage_tile_async(
    const float* __restrict__ trainX, int row0, unsigned lds_tile_base, int lane) {
  const int r = lane & 15;
  const int h = lane >> 4;
  const float* gsrc = trainX + (size_t)(row0 + r) * DIM + 2 * h;
#pragma unroll
  for (int kk = 0; kk < 16; ++kk) {
    unsigned loff = lds_tile_base + (unsigned)((kk * 32 + lane) * 8);
    const void* g = (const void*)(gsrc + 4 * kk);
    asm volatile("global_load_async_to_lds_b64 %0, %1, off"
                 :: "v"(loff), "v"(g) : "memory");
  }
}

// ---------------------------------------------------------------- main fused kernel
__global__ __launch_bounds__(WPB * 32) void krr_main_kernel(
    const float* __restrict__ testX, const float* __restrict__ trainX,
    const float* __restrict__ alphas, const float* __restrict__ ls,
    const float* __restrict__ xnorms, const float* __restrict__ tnorms,
    float* __restrict__ partial) {
  __shared__ __align__(16) float sbuf[2 * BUF_FLOATS];   // 64 KB double buffer

  const int lane  = threadIdx.x & 31;
  const int wave  = threadIdx.x >> 5;
  const int mtile = blockIdx.x * WPB + wave;
  const int m0    = mtile * 16;
  const int split = blockIdx.y;
  const int nlen  = NTRAIN / NSPLIT;
  const int n_beg = split * nlen;
  const int R     = nlen / ROWS_PER_ROUND;

  const int r = lane & 15;
  const int h = lane >> 4;

  const float l   = ls[0];
  const float il2 = 1.0f / (l * l);
  const float m2i = -2.0f * il2;
  const float c1  = -0.72134752044f;                     // -0.5 * log2(e)

  // low 32 bits of the flat shared-memory address == LDS byte address
  const unsigned lds_base = (unsigned)(size_t)(const void*)&sbuf[0];

  // ---- A tile: test rows m0..m0+15, all 64 K values (WMMA stripe), loaded once
  v2f a[16];
  const float* arow = testX + (size_t)(m0 + r) * DIM + 2 * h;
#pragma unroll
  for (int kk = 0; kk < 16; ++kk)
    a[kk] = *(const v2f*)(arow + 4 * kk);

  float xn[8];
#pragma unroll
  for (int j = 0; j < 8; ++j) xn[j] = xnorms[m0 + 8 * h + j] * il2;

  float acc[8];
#pragma unroll
  for (int j = 0; j < 8; ++j) acc[j] = 0.f;

  // ---- prologue: stage round 0 into buffer 0 (wave w stages tile w)
  stage_tile_async(trainX, n_beg + wave * 16, lds_base + wave * TILE_BYTES, lane);
  asm volatile("s_wait_asynccnt 0x0" ::: "memory");
  __syncthreads();

  for (int rr = 0; rr < R; ++rr) {
    const int cur = rr & 1;
    const int n0r = n_beg + rr * ROWS_PER_ROUND;

    // stage next round into the other buffer (overlaps with this round's math)
    if (rr + 1 < R)
      stage_tile_async(trainX, n_beg + (rr + 1) * ROWS_PER_ROUND + wave * 16,
                       lds_base + (cur ^ 1) * BUF_BYTES + wave * TILE_BYTES, lane);

    // batch-load this lane's alpha / train-norm for all NB tiles of the round
    float an8[NB], tn8[NB];
#pragma unroll
    for (int t = 0; t < NB; ++t) {
      an8[t] = alphas[n0r + t * 16 + r];
      tn8[t] = tnorms[n0r + t * 16 + r] * il2;
    }

    const float* lbuf = &sbuf[cur * BUF_FLOATS];
#pragma unroll
    for (int t = 0; t < NB; ++t) {
      // bank-conflict-free ds_load_b64: 32 lanes * 8B = all 64 LDS banks
      const v2f* bsrc = (const v2f*)&lbuf[t * TILE_FLOATS];
      v2f b[16];
#pragma unroll
      for (int kk = 0; kk < 16; ++kk) b[kk] = bsrc[kk * 32 + lane];

      v8f c = {};
#pragma unroll
      for (int kk = 0; kk < 16; ++kk)
        c = __builtin_amdgcn_wmma_f32_16x16x4_f32(
            false, a[kk], false, b[kk], (short)0, c, false, false);

#pragma unroll
      for (int j = 0; j < 8; ++j) {
        float d2 = fmaf(m2i, c[j], xn[j] + tn8[t]);      // scaled squared distance
        d2 = fmaxf(d2, 0.0f);
        acc[j] = fmaf(__builtin_amdgcn_exp2f(c1 * d2), an8[t], acc[j]);
      }
    }

    // next-round staging done + everyone finished reading the current buffer
    asm volatile("s_wait_asynccnt 0x0" ::: "memory");
    __syncthreads();
  }

  // ---- reduce over the 16 train columns (lanes 0-15 / 16-31 independently)
#pragma unroll
  for (int j = 0; j < 8; ++j) {
    float v = acc[j];
#pragma unroll
    for (int mask = 8; mask >= 1; mask >>= 1)
      v += __shfl_xor(v, mask, 32);
    acc[j] = v;
  }
  if (r == 0) {
#pragma unroll
    for (int j = 0; j < 8; ++j)
      partial[(size_t)split * NTEST + m0 + 8 * h + j] = acc[j];
  }
}

// ---------------------------------------------------------------- deterministic partial reduce
__global__ __launch_bounds__(256) void krr_reduce_kernel(
    const float* __restrict__ partial, float* __restrict__ out) {
  int m = blockIdx.x * blockDim.x + threadIdx.x;
  if (m >= NTEST) return;
  float s = 0.f;
#pragma unroll
  for (int sp = 0; sp < NSPLIT; ++sp) s += partial[(size_t)sp * NTEST + m];
  out[m] = s;
}

// ----------------------------------------------------------------
extern "C" void kernel_launch(void* const* d_in, const int* in_sizes, int n_in,
                              void* d_out, int out_size, void* d_ws, size_t ws_size,
                              hipStream_t stream) {
  const float* testX  = (const float*)d_in[0];
  const float* trainX = (const float*)d_in[1];
  const float* alphas = (const float*)d_in[2];
  const float* ls     = (const float*)d_in[3];
  float* out = (float*)d_out;

  float* xnorms  = (float*)d_ws;               // NTEST floats
  float* tnorms  = xnorms + NTEST;             // NTRAIN floats
  float* partial = tnorms + NTRAIN;            // NSPLIT * NTEST floats

  krr_norms_kernel<<<(NTEST + 255) / 256, 256, 0, stream>>>(testX, xnorms, NTEST);
  krr_norms_kernel<<<(NTRAIN + 255) / 256, 256, 0, stream>>>(trainX, tnorms, NTRAIN);

  dim3 grid(NTEST / 16 / WPB, NSPLIT);
  krr_main_kernel<<<grid, WPB * 32, 0, stream>>>(testX, trainX, alphas, ls,
                                                 xnorms, tnorms, partial);

  krr_reduce_kernel<<<(NTEST + 255) / 256, 256, 0, stream>>>(partial, out);
}